// Modified_SmoothL1Loss_13975823581342
// MI455X (gfx1250) — compile-verified
//
#include <hip/hip_runtime.h>
#include <hip/hip_bf16.h>

// SmoothL1(beta=0.01) sum-reduction over 16384x4096 fp32, divided by 16384.
// Bandwidth-bound: 537 MB @ 23.3 TB/s => ~23 us floor. Strategy:
//  - b128 non-temporal loads (no reuse; working set > 192 MB L2)
//  - register accumulation, wave reduction via V_WMMA_F32_16X16X4_F32
//    (B = ones => one matrix op sums the lane partials), deterministic
//    two-kernel tree (no float atomics).

typedef __attribute__((ext_vector_type(2))) float v2f;
typedef __attribute__((ext_vector_type(4))) float v4f;
typedef __attribute__((ext_vector_type(8))) float v8f;

#define BETA 0.01f
#define NBLOCKS 2048
#define NTHREADS 256
#define NWAVES (NTHREADS / 32)

__device__ __forceinline__ float smooth_l1(float xv, float yv) {
    float z  = xv - yv;
    float az = __builtin_fabsf(z);
    float q  = 0.5f * z * z;
    float l  = BETA * (az - 0.5f * BETA);
    return (az < BETA) ? q : l;
}

// Sum `acc` across all 32 lanes of the wave using the matrix engine.
// A (16x4 f32): lanes 0-15 supply K=0,1; lanes 16-31 supply K=2,3.
// With per-lane A = {acc, 0} and B = ones, D[m][*] = acc(m) + acc(m+16).
// Per-lane sum of D's 8 VGPRs covers rows 0-7 (lanes 0-15) or 8-15
// (lanes 16-31); one xor-16 exchange finishes the 32-lane total.
// PRECONDITION: EXEC all ones (call from fully active code).
__device__ __forceinline__ float wave_reduce_wmma(float acc) {
    v2f a; a.x = acc; a.y = 0.0f;
    v2f b; b.x = 1.0f; b.y = 1.0f;
    v8f c = {};
    v8f d = __builtin_amdgcn_wmma_f32_16x16x4_f32(
        /*neg_a=*/false, a, /*neg_b=*/false, b,
        /*c_mod=*/(short)0, c, /*reuse_a=*/false, /*reuse_b=*/false);
    float s = ((d[0] + d[1]) + (d[2] + d[3])) + ((d[4] + d[5]) + (d[6] + d[7]));
    s += __shfl_xor(s, 16, 32);
    return s;  // full-wave sum, present in every lane
}

__global__ void __launch_bounds__(NTHREADS)
smoothl1_partial_kernel(const float* __restrict__ x,
                        const float* __restrict__ y,
                        float* __restrict__ partials,
                        int n4, int rem) {
    const v4f* x4 = (const v4f*)x;
    const v4f* y4 = (const v4f*)y;

    float acc = 0.0f;
    int stride = gridDim.x * blockDim.x;
    #pragma unroll 4
    for (int i = blockIdx.x * blockDim.x + threadIdx.x; i < n4; i += stride) {
        v4f xv = __builtin_nontemporal_load(x4 + i);   // global_load_b128, TH=NT
        v4f yv = __builtin_nontemporal_load(y4 + i);
        acc += smooth_l1(xv.x, yv.x);
        acc += smooth_l1(xv.y, yv.y);
        acc += smooth_l1(xv.z, yv.z);
        acc += smooth_l1(xv.w, yv.w);
    }
    // Scalar tail (N % 4), handled once, deterministically.
    if (blockIdx.x == 0 && threadIdx.x == 0) {
        int base = n4 * 4;
        for (int t = 0; t < rem; ++t)
            acc += smooth_l1(x[base + t], y[base + t]);
    }

    // Wave reduction (EXEC all ones here: no divergence wraps this call).
    float wsum = wave_reduce_wmma(acc);

    __shared__ float lds[NWAVES];
    int lane = threadIdx.x & 31;
    int wave = threadIdx.x >> 5;
    if (lane == 0) lds[wave] = wsum;
    __syncthreads();

    if (threadIdx.x == 0) {
        float t = 0.0f;
        #pragma unroll
        for (int w = 0; w < NWAVES; ++w) t += lds[w];  // fixed order
        partials[blockIdx.x] = t;
    }
}

__global__ void __launch_bounds__(NTHREADS)
smoothl1_final_kernel(const float* __restrict__ partials,
                      int npartials,
                      float* __restrict__ out,
                      float inv_batch) {
    float acc = 0.0f;
    for (int i = threadIdx.x; i < npartials; i += NTHREADS)  // fixed order
        acc += partials[i];

    float wsum = wave_reduce_wmma(acc);

    __shared__ float lds[NWAVES];
    int lane = threadIdx.x & 31;
    int wave = threadIdx.x >> 5;
    if (lane == 0) lds[wave] = wsum;
    __syncthreads();

    if (threadIdx.x == 0) {
        float t = 0.0f;
        #pragma unroll
        for (int w = 0; w < NWAVES; ++w) t += lds[w];
        out[0] = t * inv_batch;
    }
}

extern "C" void kernel_launch(void* const* d_in, const int* in_sizes, int n_in,
                              void* d_out, int out_size, void* d_ws, size_t ws_size,
                              hipStream_t stream) {
    const float* x = (const float*)d_in[0];
    const float* y = (const float*)d_in[1];
    float* out = (float*)d_out;
    float* partials = (float*)d_ws;   // NBLOCKS * 4 = 8 KB scratch

    int n   = in_sizes[0];            // 16384 * 4096
    int n4  = n >> 2;
    int rem = n & 3;

    // Reference divides by batch size z.shape[0] == 16384 (shape fixed in ref).
    const float inv_batch = 1.0f / 16384.0f;

    smoothl1_partial_kernel<<<NBLOCKS, NTHREADS, 0, stream>>>(x, y, partials, n4, rem);
    smoothl1_final_kernel<<<1, NTHREADS, 0, stream>>>(partials, NBLOCKS, out, inv_batch);
}